// Loss_reference_45938970198438
// MI455X (gfx1250) — compile-verified
//
#include <hip/hip_runtime.h>

// ---------------- problem constants (fixed by setup_inputs) ----------------
constexpr int B_ = 128, L_ = 128, D_ = 300, K_ = 10, R_ = 50;
constexpr int NITER_ = 20;
#define EPS_   0.0025f      // blur^p = 0.05^2
#define IEPS_  400.0f       // 1/EPS
#define NEGW_  (-1.0e9f)    // log-weight of padded points

typedef float v2f  __attribute__((ext_vector_type(2)));
typedef float v8f  __attribute__((ext_vector_type(8)));

// ---------------- shared-memory carving ----------------
struct Smem {
    float *C, *f, *g, *ax2, *ty2, *loga, *wa, *logb, *wb, *red;
};
template<int NR, int NC, int LDC>
__device__ inline Smem carve(float* s) {
    Smem m;
    m.C   = s; s += NR * LDC;
    m.f   = s; s += NR;
    m.g   = s; s += NC;
    m.ax2 = s; s += NR;
    m.ty2 = s; s += NC;
    m.loga= s; s += NR;
    m.wa  = s; s += NR;
    m.logb= s; s += NC;
    m.wb  = s; s += NC;
    m.red = s;
    return m;
}
constexpr size_t smemBytes(int NR, int NC, int LDC) {
    return (size_t)(NR * LDC + 4 * NR + 4 * NC + 256) * sizeof(float);
}

// ---- WMMA row-of-tiles: acc[nt][m][n] += sum_d A[m0+m][d]*Bt[n0+16nt+n][d] -
// Unconditional loads: out-of-range rows are CLAMPED (their products are
// garbage-but-finite; those C entries are masked by NEG log-weights in the
// Sinkhorn logsumexp, so no zeroing / exec-predication is needed).
template<int DD, int NTW>
__device__ inline void mma_row(const float* __restrict__ A, int m0, int mValid,
                               const float* __restrict__ Bt, int n0, int nValid,
                               v8f acc[NTW])
{
    const int lane  = threadIdx.x & 31;
    const int lhalf = lane >> 4;      // 0: K pair {d,d+1}, 1: K pair {d+2,d+3}
    const int r     = lane & 15;

    int am = m0 + r; am = (am < mValid) ? am : (mValid - 1);
    const float* ap = A + (size_t)am * DD + 2 * lhalf;

    const float* bp[NTW];
    #pragma unroll
    for (int nt = 0; nt < NTW; ++nt) {
        int bn = n0 + nt * 16 + r; bn = (bn < nValid) ? bn : (nValid - 1);
        bp[nt] = Bt + (size_t)bn * DD + 2 * lhalf;
    }

    for (int d = 0; d < DD; d += 4) {
        v2f a = *(const v2f*)(ap + d);          // A frag reused for NTW tiles
        #pragma unroll
        for (int nt = 0; nt < NTW; ++nt) {
            v2f b = *(const v2f*)(bp[nt] + d);
            acc[nt] = __builtin_amdgcn_wmma_f32_16x16x4_f32(
                false, a, false, b, (short)0, acc[nt], false, false);
        }
    }
}

// ---------------- build C[m][n] = 0.5*max(|x|^2+|y|^2-2 x.y, 0) in LDS ------
template<int NR, int NC, int LDC, int DD>
__device__ inline void build_cost(const float* __restrict__ A, int mValid,
                                  const float* __restrict__ Bt, int nValid,
                                  const Smem& sm)
{
    const int tid = threadIdx.x;
    const int wv = tid >> 5, lane = tid & 31;
    const int lhalf = lane >> 4, r = lane & 15;
    constexpr int MT = NR / 16, NT = NC / 16;
    constexpr int NTW = (MT * NT) / 8;            // n-tiles per wave
    static_assert(MT * NT == NTW * 8, "8 waves cover all tiles");

    const int mtile = wv % MT;
    const int nt0   = (wv / MT) * NTW;

    v8f acc[NTW];
    const v8f vzero = {};
    #pragma unroll
    for (int nt = 0; nt < NTW; ++nt) acc[nt] = vzero;

    mma_row<DD, NTW>(A, mtile * 16, mValid, Bt, nt0 * 16, nValid, acc);

    // C/D layout: VGPR j, lanes 0-15 -> M=j, lanes 16-31 -> M=j+8
    #pragma unroll
    for (int nt = 0; nt < NTW; ++nt) {
        const int n = (nt0 + nt) * 16 + r;
        #pragma unroll
        for (int jj = 0; jj < 8; ++jj) {
            int m = mtile * 16 + lhalf * 8 + jj;
            float c = 0.5f * fmaxf(sm.ax2[m] + sm.ty2[n] - 2.0f * acc[nt][jj], 0.f);
            sm.C[m * LDC + n] = c;
        }
    }
}

// ---------------- log-domain Sinkhorn, fully in LDS ----------------
// f[m] = -eps * lse_n( logb[n] + (g[n]-C[m,n])/eps )   then
// g[n] = -eps * lse_m( loga[m] + (f[m]-C[m,n])/eps )
template<int NR, int NC, int LDC, int TPR, int TPC>
__device__ inline void sinkhorn_iters(const Smem& sm, int tid)
{
    static_assert(NR * TPR == 256 && NC * TPC == 256, "block = 256");
    for (int it = 0; it < NITER_; ++it) {
        {   // ---- f update: TPR lanes cooperate per row ----
            int m = tid / TPR, q = tid % TPR;
            float mx = -3.402823466e38f, s = 0.f;
            for (int n = q; n < NC; n += TPR) {
                float z = sm.logb[n] + (sm.g[n] - sm.C[m * LDC + n]) * IEPS_;
                if (z > mx) { s = s * __expf(mx - z) + 1.f; mx = z; }
                else        { s += __expf(z - mx); }
            }
            #pragma unroll
            for (int off = 1; off < TPR; off <<= 1) {
                float mx2 = __shfl_xor(mx, off, 32);
                float s2  = __shfl_xor(s,  off, 32);
                float M = fmaxf(mx, mx2);
                s = s * __expf(mx - M) + s2 * __expf(mx2 - M);
                mx = M;
            }
            if (q == 0) sm.f[m] = -EPS_ * (mx + __logf(s));
            __syncthreads();
        }
        {   // ---- g update: TPC lanes cooperate per column ----
            int n = tid / TPC, q = tid % TPC;
            float mx = -3.402823466e38f, s = 0.f;
            for (int m = q; m < NR; m += TPC) {
                float z = sm.loga[m] + (sm.f[m] - sm.C[m * LDC + n]) * IEPS_;
                if (z > mx) { s = s * __expf(mx - z) + 1.f; mx = z; }
                else        { s += __expf(z - mx); }
            }
            #pragma unroll
            for (int off = 1; off < TPC; off <<= 1) {
                float mx2 = __shfl_xor(mx, off, 32);
                float s2  = __shfl_xor(s,  off, 32);
                float M = fmaxf(mx, mx2);
                s = s * __expf(mx - M) + s2 * __expf(mx2 - M);
                mx = M;
            }
            if (q == 0) sm.g[n] = -EPS_ * (mx + __logf(s));
            __syncthreads();
        }
    }
}

// ---------------- OT value = <a, f> + <b, g>, block reduce ----------------
__device__ inline void ot_reduce_store(const Smem& sm, int NRv, int NCv,
                                       float* out, int tid)
{
    float val = 0.f;
    if (tid < NRv) val += sm.wa[tid] * sm.f[tid];
    if (tid < NCv) val += sm.wb[tid] * sm.g[tid];
    sm.red[tid] = val;
    __syncthreads();
    for (int s = 128; s > 0; s >>= 1) {
        if (tid < s) sm.red[tid] += sm.red[tid + s];
        __syncthreads();
    }
    if (tid == 0) *out = sm.red[0];
}

// ================= kernel 1: ot_tt[i,j], 50x50 (padded to 64x64) ===========
__global__ __launch_bounds__(256) void ot_tt_kernel(const float* __restrict__ t0,
                                                    float* __restrict__ out)
{
    constexpr int NR = 64, NC = 64, LDC = 65, TPR = 4, TPC = 4;
    extern __shared__ float smem[];
    Smem sm = carve<NR, NC, LDC>(smem);
    const int tid = threadIdx.x;
    const int i = blockIdx.x / K_, j = blockIdx.x % K_;
    const float* A  = t0 + (size_t)i * R_ * D_;
    const float* Bt = t0 + (size_t)j * R_ * D_;
    const float logr = -__logf((float)R_);

    if (tid < NR) {
        float s = 0.f;
        if (tid < R_) { const float* p = A + (size_t)tid * D_;
                        for (int d = 0; d < D_; ++d) { float v = p[d]; s += v * v; } }
        sm.ax2[tid] = s;
        sm.loga[tid] = (tid < R_) ? logr : NEGW_;
        sm.wa[tid]   = (tid < R_) ? 1.f / (float)R_ : 0.f;
        sm.f[tid] = 0.f;
    } else if (tid < NR + NC) {
        int n = tid - NR; float s = 0.f;
        if (n < R_) { const float* p = Bt + (size_t)n * D_;
                      for (int d = 0; d < D_; ++d) { float v = p[d]; s += v * v; } }
        sm.ty2[n] = s;
        sm.logb[n] = (n < R_) ? logr : NEGW_;
        sm.wb[n]   = (n < R_) ? 1.f / (float)R_ : 0.f;
        sm.g[n] = 0.f;
    }
    __syncthreads();
    build_cost<NR, NC, LDC, D_>(A, R_, Bt, R_, sm);
    __syncthreads();
    sinkhorn_iters<NR, NC, LDC, TPR, TPC>(sm, tid);
    ot_reduce_store(sm, NR, NC, out + blockIdx.x, tid);
}

// ================= kernel 2: ot_ab[b,k], 128x50 (padded to 128x64) =========
__global__ __launch_bounds__(256) void ot_ab_kernel(const float* __restrict__ anchor,
                                                    const float* __restrict__ weight,
                                                    const float* __restrict__ t0,
                                                    const int* __restrict__ lenA,
                                                    float* __restrict__ out)
{
    constexpr int NR = 128, NC = 64, LDC = 65, TPR = 2, TPC = 4;
    extern __shared__ float smem[];
    Smem sm = carve<NR, NC, LDC>(smem);
    const int tid = threadIdx.x;
    const int b = blockIdx.x / K_, k = blockIdx.x % K_;
    const float* A  = anchor + (size_t)b * L_ * D_;
    const float* Bt = t0 + (size_t)k * R_ * D_;
    const int len = lenA[b];

    if (tid < NR) {
        const float* p = A + (size_t)tid * D_;
        float s = 0.f;
        for (int d = 0; d < D_; ++d) { float v = p[d]; s += v * v; }
        sm.ax2[tid] = s;
        float w = fmaxf(weight[(size_t)b * L_ + tid], 1e-12f);
        bool valid = tid < len;
        sm.loga[tid] = valid ? __logf(w) : NEGW_;
        sm.wa[tid]   = valid ? w : 0.f;
        sm.f[tid] = 0.f;
    } else if (tid < NR + NC) {
        int n = tid - NR; float s = 0.f;
        if (n < R_) { const float* p = Bt + (size_t)n * D_;
                      for (int d = 0; d < D_; ++d) { float v = p[d]; s += v * v; } }
        sm.ty2[n] = s;
        sm.logb[n] = (n < R_) ? -__logf((float)R_) : NEGW_;
        sm.wb[n]   = (n < R_) ? 1.f / (float)R_ : 0.f;
        sm.g[n] = 0.f;
    }
    __syncthreads();
    build_cost<NR, NC, LDC, D_>(A, L_, Bt, R_, sm);
    __syncthreads();
    sinkhorn_iters<NR, NC, LDC, TPR, TPC>(sm, tid);
    ot_reduce_store(sm, NR, NC, out + blockIdx.x, tid);
}

// ================= kernel 3: ot_aa[b], 128x128 =============================
__global__ __launch_bounds__(256) void ot_aa_kernel(const float* __restrict__ anchor,
                                                    const float* __restrict__ weight,
                                                    const int* __restrict__ lenA,
                                                    float* __restrict__ out)
{
    constexpr int NR = 128, NC = 128, LDC = 129, TPR = 2, TPC = 2;
    extern __shared__ float smem[];
    Smem sm = carve<NR, NC, LDC>(smem);
    const int tid = threadIdx.x;
    const int b = blockIdx.x;
    const float* A = anchor + (size_t)b * L_ * D_;
    const int len = lenA[b];

    if (tid < NR) {
        const float* p = A + (size_t)tid * D_;
        float s = 0.f;
        for (int d = 0; d < D_; ++d) { float v = p[d]; s += v * v; }
        float w = fmaxf(weight[(size_t)b * L_ + tid], 1e-12f);
        bool valid = tid < len;
        float lw = valid ? __logf(w) : NEGW_;
        float we = valid ? w : 0.f;
        sm.ax2[tid] = s;  sm.ty2[tid]  = s;   // symmetric problem
        sm.loga[tid] = lw; sm.logb[tid] = lw;
        sm.wa[tid]  = we; sm.wb[tid]   = we;
        sm.f[tid] = 0.f;  sm.g[tid] = 0.f;
    }
    __syncthreads();
    build_cost<NR, NC, LDC, D_>(A, L_, A, L_, sm);
    __syncthreads();
    sinkhorn_iters<NR, NC, LDC, TPR, TPC>(sm, tid);
    ot_reduce_store(sm, NR, NC, out + blockIdx.x, tid);
}

// ================= kernel 4: debiased divergences + margin loss ============
__global__ __launch_bounds__(256) void finalize_kernel(const float* __restrict__ tt,
                                                       const float* __restrict__ ab,
                                                       const float* __restrict__ aa,
                                                       const int* __restrict__ grade,
                                                       float* __restrict__ out)
{
    __shared__ float selfT[K_];
    __shared__ float red[256];
    const int tid = threadIdx.x;
    if (tid < K_) selfT[tid] = tt[tid * K_ + tid];
    __syncthreads();

    float disPart = 0.f;
    if (tid < K_ * K_) disPart += tt[tid];
    if (tid < K_)      disPart -= (float)K_ * selfT[tid];

    float lossPart = 0.f;
    if (tid < B_) {
        const int b = tid, g = grade[b];
        const float aav = aa[b];
        float pos = 0.f;
        for (int k = 0; k < K_; ++k) {
            float dk = ab[b * K_ + k] - 0.5f * (aav + selfT[k]);
            if (k == g) pos = dk;
        }
        float lb = 0.f;
        for (int k = 0; k < K_; ++k) {
            float dk = ab[b * K_ + k] - 0.5f * (aav + selfT[k]);
            lb += fmaxf(pos - dk + 10.0f, 0.f);
        }
        lossPart = lb - 10.0f;
    }
    red[tid] = lossPart * (1.0f / (float)B_) - disPart * 0.01f;
    __syncthreads();
    for (int s = 128; s > 0; s >>= 1) {
        if (tid < s) red[tid] += red[tid + s];
        __syncthreads();
    }
    if (tid == 0) out[0] = red[0];
}

// ================= host entry ==============================================
extern "C" void kernel_launch(void* const* d_in, const int* in_sizes, int n_in,
                              void* d_out, int out_size, void* d_ws, size_t ws_size,
                              hipStream_t stream)
{
    const float* anchor = (const float*)d_in[0];   // [B, L, D]
    const float* weight = (const float*)d_in[1];   // [B, L]
    const float* t0     = (const float*)d_in[2];   // [K, R, D]
    const int*   lenA   = (const int*)d_in[3];     // [B]
    const int*   grade  = (const int*)d_in[4];     // [B]
    (void)in_sizes; (void)n_in; (void)out_size; (void)ws_size;

    float* ws = (float*)d_ws;
    float* tt = ws;                 // K*K = 100 (pad to 128)
    float* ab = ws + 128;           // B*K = 1280
    float* aa = ws + 128 + 1280;    // B   = 128

    // aa kernel needs ~70KB dynamic LDS (fine on CDNA5: 320KB/WGP)
    (void)hipFuncSetAttribute((const void*)ot_aa_kernel,
                              hipFuncAttributeMaxDynamicSharedMemorySize,
                              (int)smemBytes(128, 128, 129));

    ot_tt_kernel<<<K_ * K_, 256, smemBytes(64, 64, 65),   stream>>>(t0, tt);
    ot_ab_kernel<<<B_ * K_, 256, smemBytes(128, 64, 65),  stream>>>(anchor, weight, t0, lenA, ab);
    ot_aa_kernel<<<B_,      256, smemBytes(128, 128, 129), stream>>>(anchor, weight, lenA, aa);
    finalize_kernel<<<1, 256, 0, stream>>>(tt, ab, aa, grade, (float*)d_out);
}